// SpatialEMOSprior_34033320853735
// MI455X (gfx1250) — compile-verified
//
#include <hip/hip_runtime.h>
#include <math.h>

// ---------------------------------------------------------------------------
// SpatialEMOS prior head for MI455X (gfx1250, wave32).
//   theta[b,m,pix] = sum_i feat[b,i,pix]*mat[m,i,pix]   (F=64, M=4)
//   mu_pred    = mu*(th0+1)+th1
//   sigma_pred = exp(log(sigma)*(th2+1)+th3)
// Memory-bound (~1.75 FLOP/B); contraction mapped to V_WMMA_F32_16X16X4_F32:
// per pixel, theta(16b x 4m) = A(16x64) @ B(64x4), K chained in 16 steps,
// B N-dim zero-padded 4->16. Global loads stay pixel-coalesced; LDS does the
// lane-layout transpose for the WMMA operands.
// ---------------------------------------------------------------------------

typedef __attribute__((ext_vector_type(2))) float v2f;
typedef __attribute__((ext_vector_type(8))) float v8f;

#define NPIX   29040          // 121*240
#define NB     64
#define NF     64
#define NM     4
#define PT     16             // pixels per workgroup (NPIX/PT = 1815 exact)
#define BT     16             // batch tile = WMMA M
#define NKS    16             // K steps: 64/4
#define FSTR   (NF*PT + 2)    // 1026 floats: 8B-aligned rows, bank stride 2 -> no conflicts
#define MSTR   (NF*PT + 2)
#define THREADS 512           // 16 waves, one pixel per wave

__global__ __launch_bounds__(THREADS)
void emos_wmma_kernel(const float* __restrict__ gmu,
                      const float* __restrict__ gsigma,
                      const float* __restrict__ gfeat,
                      const float* __restrict__ gmat,
                      float* __restrict__ gout)
{
    __shared__ float smF[BT * FSTR];        // features tile [b][i][pix] (padded)
    __shared__ float smM[NM * MSTR];        // matrix tile   [m][i][pix] (padded)
    __shared__ float smT[NB * NM * PT];     // theta         [b][m][pix]

    const int tid   = threadIdx.x;
    const int pix0  = blockIdx.x * PT;
    const int wv    = tid >> 5;
    const int lane  = tid & 31;
    const int rowc  = lane & 15;   // A row (b-local) == B/D column (m)
    const int hf    = lane >> 4;   // K-half select per ISA 16x4 f32 layout
    const int mypix = wv;          // one pixel per wave

    // ---- stage matrix tile (once): coalesced float4 global loads ----
    for (int c = tid; c < (NM * NF * PT) / 4; c += THREADS) {
        const int mi = c >> 2, quad = c & 3;
        const int m = mi >> 6, i = mi & 63;
        const float4 v = *(const float4*)(gmat + (size_t)(m * NF + i) * NPIX + pix0 + quad * 4);
        float* p = &smM[m * MSTR + i * PT + quad * 4];
        p[0] = v.x; p[1] = v.y; p[2] = v.z; p[3] = v.w;
    }
    __syncthreads();

    // ---- preload B operands (constant across all b tiles) into registers ----
    // Assumed B(4x16) layout (mirror of documented A 16x4):
    //   VGPR0: lanes0-15 K=k0+0, lanes16-31 K=k0+2 ; VGPR1: K=k0+1 / k0+3.
    v2f Breg[NKS];
#pragma unroll
    for (int s = 0; s < NKS; ++s) {
        const int kA = 4 * s + 2 * hf;
        float b0 = 0.0f, b1 = 0.0f;
        if (rowc < NM) {
            b0 = smM[rowc * MSTR + kA * PT + mypix];
            b1 = smM[rowc * MSTR + (kA + 1) * PT + mypix];
        }
        v2f t; t.x = b0; t.y = b1;
        Breg[s] = t;
    }

    // ---- loop over 4 b-tiles of 16 ----
    for (int b0 = 0; b0 < NB; b0 += BT) {
        __syncthreads();   // previous tile's compute done before smF overwrite
        for (int c = tid; c < (BT * NF * PT) / 4; c += THREADS) {
            const int bi = c >> 2, quad = c & 3;
            const int bl = bi >> 6, i = bi & 63;
            const float4 v = *(const float4*)(gfeat + (size_t)((b0 + bl) * NF + i) * NPIX + pix0 + quad * 4);
            float* p = &smF[bl * FSTR + i * PT + quad * 4];
            p[0] = v.x; p[1] = v.y; p[2] = v.z; p[3] = v.w;
        }
        if (b0 + BT < NB) {  // light prefetch of next tile -> global_prefetch_b8
            const int bi = tid >> 2, quad = tid & 3;
            const int bl = bi >> 6, i = bi & 63;
            __builtin_prefetch(gfeat + (size_t)((b0 + BT + bl) * NF + i) * NPIX + pix0 + quad * 4, 0, 0);
        }
        __syncthreads();

        // ---- 16-step K chain: theta(16x16) += A(16x4) x B(4x16) ----
        v8f acc = {0.f, 0.f, 0.f, 0.f, 0.f, 0.f, 0.f, 0.f};
#pragma unroll
        for (int s = 0; s < NKS; ++s) {
            const int kA = 4 * s + 2 * hf;
            v2f a;
            a.x = smF[rowc * FSTR + kA * PT + mypix];        // A[M=rowc][K=kA]
            a.y = smF[rowc * FSTR + (kA + 1) * PT + mypix];  // A[M=rowc][K=kA+1]
            acc = __builtin_amdgcn_wmma_f32_16x16x4_f32(false, a, false, Breg[s],
                                                        (short)0, acc, false, false);
        }

        // D lane layout: VGPR r at lane l = D[r + 8*(l/16)][l%16]; keep m = rowc < 4
        if (rowc < NM) {
#pragma unroll
            for (int r = 0; r < 8; ++r) {
                const int bg = b0 + r + 8 * hf;
                smT[bg * (NM * PT) + rowc * PT + mypix] = acc[r];
            }
        }
    }
    __syncthreads();

    // ---- fused finalize: coalesced over (b, pix) ----
    const size_t outN = (size_t)NB * NPIX;
    for (int e = tid; e < NB * PT; e += THREADS) {
        const int b = e >> 4, px = e & 15;
        const size_t g = (size_t)b * NPIX + pix0 + px;
        const float th0 = smT[b * (NM * PT) + 0 * PT + px];
        const float th1 = smT[b * (NM * PT) + 1 * PT + px];
        const float th2 = smT[b * (NM * PT) + 2 * PT + px];
        const float th3 = smT[b * (NM * PT) + 3 * PT + px];
        const float muv = gmu[g];
        const float sgv = gsigma[g];
        gout[g]        = muv * (th0 + 1.0f) + th1;
        gout[outN + g] = expf(logf(sgv) * (th2 + 1.0f) + th3);
    }
}

extern "C" void kernel_launch(void* const* d_in, const int* in_sizes, int n_in,
                              void* d_out, int out_size, void* d_ws, size_t ws_size,
                              hipStream_t stream) {
    // setup_inputs order: mu, sigma, features, truth, matrix (all float32)
    const float* gmu    = (const float*)d_in[0];
    const float* gsigma = (const float*)d_in[1];
    const float* gfeat  = (const float*)d_in[2];
    // d_in[3] = truth (unused by reference forward)
    const float* gmat   = (const float*)d_in[4];
    float* gout = (float*)d_out;   // [mu_pred | sigma_pred], each NB*NPIX floats

    (void)in_sizes; (void)n_in; (void)out_size; (void)d_ws; (void)ws_size;

    dim3 grid(NPIX / PT);          // 1815 workgroups
    dim3 block(THREADS);           // 16 wave32s
    emos_wmma_kernel<<<grid, block, 0, stream>>>(gmu, gsigma, gfeat, gmat, gout);
}